// SpGAT_2_86887188398710
// MI455X (gfx1250) — compile-verified
//
#include <hip/hip_runtime.h>
#include <hip/hip_bf16.h>

// ---------------- problem constants (match reference) ----------------
static constexpr int NN     = 50000;    // nodes
static constexpr int NFEAT  = 256;
static constexpr int NHID   = 64;
static constexpr int NHEADS = 8;
static constexpr int NCLASS = 40;
static constexpr int NSTRUC = 128;
static constexpr int ECNT   = 1600000;  // edges (divisible by 32)
static constexpr int DCAT   = NHID * NHEADS;  // 512
static constexpr float ALPHA = 0.2f;

typedef __attribute__((ext_vector_type(2))) float v2f;
typedef __attribute__((ext_vector_type(8))) float v8f;

// ---------------- small device helpers ----------------
// Native CDNA5 f32 atomic add, no return (STOREcnt-tracked). We never read the
// accumulator in the same kernel and S_ENDPGM does an implicit wait-idle.
__device__ __forceinline__ void atomAddF(float* p, float v) {
  asm volatile("global_atomic_add_f32 %0, %1, off" : : "v"(p), "v"(v) : "memory");
}
__device__ __forceinline__ float wsum(float v) {
#pragma unroll
  for (int m = 16; m; m >>= 1) v += __shfl_xor(v, m, 32);
  return v;
}
__device__ __forceinline__ float wmaxr(float v) {
#pragma unroll
  for (int m = 16; m; m >>= 1) v = fmaxf(v, __shfl_xor(v, m, 32));
  return v;
}
__device__ __forceinline__ float eluf(float v) { return v > 0.f ? v : __expf(v) - 1.f; }
__device__ __forceinline__ v8f wmma4(v2f a, v2f b, v8f c) {
  return __builtin_amdgcn_wmma_f32_16x16x4_f32(false, a, false, b, (short)0, c, false, false);
}

// ---------------- weight repack: W_heads[8][256][64] -> Wpk[256][512] ----------------
__global__ void repack_W(const float* __restrict__ Wh, float* __restrict__ Wpk) {
  int t = blockIdx.x * blockDim.x + threadIdx.x;
  if (t >= NHEADS * NFEAT * NHID) return;
  int h = t >> 14;            // / (256*64)
  int r = t & 16383;
  int k = r >> 6;             // feature row
  int j = r & 63;             // hid col
  Wpk[(size_t)k * DCAT + h * NHID + j] = Wh[t];
}

// ---------------- pad W_out[512][40] into Wop[512][64] (Wop pre-zeroed) -------------
__global__ void pad_Wout(const float* __restrict__ Wo, float* __restrict__ Wop) {
  int t = blockIdx.x * blockDim.x + threadIdx.x;
  if (t >= DCAT * NCLASS) return;
  int k = t / NCLASS, c = t % NCLASS;
  Wop[(size_t)k * 64 + c] = Wo[t];
}

// ---------------- transpose src[R][C] -> dst[C][R] ----------------
__global__ void transpose_kn(const float* __restrict__ src, float* __restrict__ dst,
                             int R, int C) {
  int t = blockIdx.x * blockDim.x + threadIdx.x;
  if (t >= R * C) return;
  int r = t / C, c = t % C;
  dst[(size_t)c * R + r] = src[t];
}

// ---------------- pack row-major B[K][NnPad] into WMMA fragment order ----------------
// For K-group g (k = 4g..4g+3), lane-half hh (0: k=4g,4g+1 ; 1: k=4g+2,4g+3), col c:
//   Bp[float2 index (g*2+hh)*NnPad + c] = ( B[4g+2hh][c], B[4g+2hh+1][c] )
// so the GEMM's per-lane B fragment is a single coalesced global_load_b64.
__global__ void pack_frag(const float* __restrict__ Bsrc, float* __restrict__ Bp,
                          int K, int NnPad) {
  int t = blockIdx.x * blockDim.x + threadIdx.x;
  if (t >= K * NnPad) return;
  int j  = t & 1;
  int r  = t >> 1;
  int c  = r % NnPad;
  int q  = r / NnPad;
  int hh = q & 1;
  int g  = q >> 1;
  int k  = 4 * g + 2 * hh + j;
  Bp[t] = Bsrc[(size_t)k * NnPad + c];
}

// ---------------- fp32 WMMA GEMM: C[M,NnReal] = A[M,K] @ Bfrag (+bias) --------------
// One wave computes a 16x64 strip: 4 accumulators, 4 WMMAs per K-step of 4.
// B is zero-padded to NnPad (multiple of 64) and fragment-packed, so the inner loop
// is 1 A b64 + 4 B b64 loads + 4 WMMAs. Requires M % 16 == 0 and K % 4 == 0.
__global__ void gemm_f32_wmma_n64(const float* __restrict__ A, const float* __restrict__ Bp,
                                  const float* __restrict__ bias, float* __restrict__ C,
                                  int M, int NnReal, int NnPad, int K, int ldc) {
  int lane = threadIdx.x & 31;
  int wid  = (blockIdx.x * blockDim.x + threadIdx.x) >> 5;
  int groupsN = NnPad >> 6;
  int tilesM  = M >> 4;
  if (wid >= tilesM * groupsN) return;
  int tm = (wid / groupsN) << 4;
  int tn = (wid % groupsN) << 6;

  int l15 = lane & 15;
  int hh  = lane >> 4;         // half-wave: 0 -> K+0,1 ; 1 -> K+2,3
  const float* arow = A + (size_t)(tm + l15) * K;
  const float2* bf  = (const float2*)Bp;

  v8f acc[4];
  acc[0] = v8f{}; acc[1] = v8f{}; acc[2] = v8f{}; acc[3] = v8f{};

  for (int kk = 0; kk < K; kk += 4) {
    float2 av = *(const float2*)(arow + kk + 2 * hh);   // 8B-aligned
    v2f a; a.x = av.x; a.y = av.y;
    const float2* bp = bf + (size_t)((kk >> 2) * 2 + hh) * NnPad + tn + l15;
    float2 b0 = bp[0], b1 = bp[16], b2 = bp[32], b3 = bp[48];
    v2f b;
    b.x = b0.x; b.y = b0.y; acc[0] = wmma4(a, b, acc[0]);
    b.x = b1.x; b.y = b1.y; acc[1] = wmma4(a, b, acc[1]);
    b.x = b2.x; b.y = b2.y; acc[2] = wmma4(a, b, acc[2]);
    b.x = b3.x; b.y = b3.y; acc[3] = wmma4(a, b, acc[3]);
  }

  int rbase = tm + (hh << 3);  // lanes 0-15 -> M=tm+r ; 16-31 -> M=tm+8+r
#pragma unroll
  for (int t = 0; t < 4; ++t) {
    int colt = tn + t * 16 + l15;
    if (colt < NnReal) {
      float bv = bias ? bias[colt] : 0.f;
#pragma unroll
      for (int r = 0; r < 8; ++r)
        C[(size_t)(rbase + r) * ldc + colt] = acc[t][r] + bv;
    }
  }
}

// ---------------- per-(node,head) attention scores: s = h_head . a ----------------
__global__ void head_scores(const float* __restrict__ h_all, const float* __restrict__ a_heads,
                            float* __restrict__ ssrc, float* __restrict__ sdst) {
  int lane = threadIdx.x & 31;
  int wid  = (blockIdx.x * blockDim.x + threadIdx.x) >> 5;
  if (wid >= NHEADS * NN) return;
  int h = wid & 7, n = wid >> 3;
  const float* hp = h_all + (size_t)n * DCAT + h * NHID;
  const float* ap = a_heads + h * 2 * NHID;
  float h0 = hp[lane], h1 = hp[lane + 32];
  float ps = h0 * ap[lane]       + h1 * ap[lane + 32];
  float pd = h0 * ap[64 + lane]  + h1 * ap[96 + lane];
  ps = wsum(ps);
  pd = wsum(pd);
  if (lane == 0) {
    ssrc[(size_t)h * NN + n] = ps;
    sdst[(size_t)h * NN + n] = pd;
  }
}

// ---------------- per-head edge aggregation: one wave owns 32 edges ------------------
// Lane i loads edge (base+i)'s indices coalesced and computes its weight (one exp per
// edge). The wave then iterates the 32 edges, broadcasting (s,d,w) with __shfl, and
// does the 64-wide gather + native f32 atomics (L2-resident per-head slices).
__global__ void edge_head(const int* __restrict__ src, const int* __restrict__ dst,
                          const float* __restrict__ ssrc, const float* __restrict__ sdst,
                          const float* __restrict__ h_all, float* __restrict__ agg,
                          float* __restrict__ rows, int head) {
  int lane = threadIdx.x & 31;
  long long wavid = ((long long)blockIdx.x * blockDim.x + threadIdx.x) >> 5;
  long long base = wavid * 32;
  if (base >= ECNT) return;               // ECNT % 32 == 0: full waves only
  long long myE = base + lane;
  int si = src[myE];
  int di = dst[myE];
  if (myE + 4096 < ECNT) __builtin_prefetch(src + myE + 4096, 0, 1);
  float e  = ssrc[si] + sdst[di];
  float lr = e > 0.f ? e : ALPHA * e;
  float wv = __expf(-lr);

  const float* hb = h_all + head * NHID + lane * 2;
  float* ab = agg + head * NHID + lane * 2;
#pragma unroll 4
  for (int i = 0; i < 32; ++i) {
    int   s = __shfl(si, i, 32);
    int   d = __shfl(di, i, 32);
    float w = __shfl(wv, i, 32);
    if (lane == 0) atomAddF(&rows[s], w);
    const float2 hv = *(const float2*)(hb + (size_t)d * DCAT);
    float* ap = ab + (size_t)s * DCAT;
    atomAddF(ap,     w * hv.x);
    atomAddF(ap + 1, w * hv.y);
  }
}

// ---------------- hp = elu(agg / rowsum) in place -> hcat (float4) ----------------
__global__ void norm_heads(float* __restrict__ agg, const float* __restrict__ rows) {
  long long t = (long long)blockIdx.x * blockDim.x + threadIdx.x;
  if (t >= (long long)NN * DCAT / 4) return;
  int n  = (int)(t >> 7);   // 128 float4 per row
  int c4 = (int)(t & 127);
  int h  = c4 >> 4;         // 16 float4 per head
  float r = rows[(size_t)h * NN + n];
  float4 v = ((float4*)agg)[t];
  v.x = eluf(v.x / r); v.y = eluf(v.y / r);
  v.z = eluf(v.z / r); v.w = eluf(v.w / r);
  ((float4*)agg)[t] = v;
}

// ---------------- output-layer per-node scores (40-dim dot, wave per node) ----------
__global__ void out_scores(const float* __restrict__ ho, const float* __restrict__ a_out,
                           float* __restrict__ ss, float* __restrict__ sd) {
  int lane = threadIdx.x & 31;
  int n = (blockIdx.x * blockDim.x + threadIdx.x) >> 5;
  if (n >= NN) return;
  const float* hp = ho + (size_t)n * NCLASS;
  float h0 = hp[lane];
  float h1 = (lane < 8) ? hp[32 + lane] : 0.f;
  float ps = h0 * a_out[lane]          + ((lane < 8) ? h1 * a_out[32 + lane] : 0.f);
  float pd = h0 * a_out[NCLASS + lane] + ((lane < 8) ? h1 * a_out[NCLASS + 32 + lane] : 0.f);
  ps = wsum(ps);
  pd = wsum(pd);
  if (lane == 0) { ss[n] = ps; sd[n] = pd; }
}

// ---------------- output-layer edge aggregation (40 dims, wave owns 32 edges) -------
__global__ void edge_out(const int* __restrict__ src, const int* __restrict__ dst,
                         const float* __restrict__ ss, const float* __restrict__ sd,
                         const float* __restrict__ ho, float* __restrict__ agg,
                         float* __restrict__ rows) {
  int lane = threadIdx.x & 31;
  long long wavid = ((long long)blockIdx.x * blockDim.x + threadIdx.x) >> 5;
  long long base = wavid * 32;
  if (base >= ECNT) return;
  long long myE = base + lane;
  int si = src[myE];
  int di = dst[myE];
  float e  = ss[si] + sd[di];
  float lr = e > 0.f ? e : ALPHA * e;
  float wv = __expf(-lr);

  bool vec = (lane < 20);
#pragma unroll 4
  for (int i = 0; i < 32; ++i) {
    int   s = __shfl(si, i, 32);
    int   d = __shfl(di, i, 32);
    float w = __shfl(wv, i, 32);
    if (lane == 0) atomAddF(&rows[s], w);
    if (vec) {
      const float2 hv = *(const float2*)(ho + (size_t)d * NCLASS + lane * 2);
      float* ap = agg + (size_t)s * NCLASS + lane * 2;
      atomAddF(ap,     w * hv.x);
      atomAddF(ap + 1, w * hv.y);
    }
  }
}

// ---------------- xo = agg_out / rowsum (no activation) in place ----------------
__global__ void norm_out(float* __restrict__ agg, const float* __restrict__ rows) {
  long long t = (long long)blockIdx.x * blockDim.x + threadIdx.x;
  if (t >= (long long)NN * NCLASS) return;
  int n = (int)(t / NCLASS);
  agg[t] = agg[t] / rows[n];
}

// ---------------- out0 = log_softmax(elu(xo)) over 40 classes (wave per node) -------
__global__ void logsoftmax40(const float* __restrict__ xo, float* __restrict__ out0) {
  int lane = threadIdx.x & 31;
  int n = (blockIdx.x * blockDim.x + threadIdx.x) >> 5;
  if (n >= NN) return;
  const float* xp = xo + (size_t)n * NCLASS;
  float v0 = eluf(xp[lane]);
  float v1 = -1e30f;
  if (lane < 8) v1 = eluf(xp[32 + lane]);
  float m = wmaxr(fmaxf(v0, v1));
  float s = __expf(v0 - m) + ((lane < 8) ? __expf(v1 - m) : 0.f);
  s = wsum(s);
  float lse = m + __logf(s);
  out0[(size_t)n * NCLASS + lane] = v0 - lse;
  if (lane < 8) out0[(size_t)n * NCLASS + 32 + lane] = v1 - lse;
}

// =====================================================================================
extern "C" void kernel_launch(void* const* d_in, const int* in_sizes, int n_in,
                              void* d_out, int out_size, void* d_ws, size_t ws_size,
                              hipStream_t stream) {
  (void)in_sizes; (void)n_in; (void)out_size; (void)ws_size;

  const float* x       = (const float*)d_in[0];
  const int*   eidx    = (const int*)d_in[1];
  const float* W_heads = (const float*)d_in[2];
  const float* a_heads = (const float*)d_in[3];
  const float* W_out   = (const float*)d_in[4];
  const float* a_out   = (const float*)d_in[5];
  const float* enc1_w  = (const float*)d_in[6];
  const float* enc1_b  = (const float*)d_in[7];
  const float* enc2_w  = (const float*)d_in[8];
  const float* enc2_b  = (const float*)d_in[9];
  const int* src = eidx;          // edge_index[0]
  const int* dst = eidx + ECNT;   // edge_index[1]

  float* out0 = (float*)d_out;                       // [N,40] log_softmax
  float* outy = out0 + (size_t)NN * NCLASS;          // [N,128] y
  float* outz = outy + (size_t)NN * NSTRUC;          // [N,128] z

  // ---- workspace carve-up (aligned 256B) ----
  char* wbase = (char*)d_ws;
  size_t off = 0;
  auto alloc = [&](size_t bytes) -> float* {
    float* p = (float*)(wbase + off);
    off += (bytes + 255) & ~(size_t)255;
    return p;
  };
  float* h_all = alloc((size_t)NN * DCAT * 4);     // x @ W_all (pre-activation)
  float* agg   = alloc((size_t)NN * DCAT * 4);     // per-head accum -> hcat in place
  float* Wpk   = alloc((size_t)NFEAT * DCAT * 4);  // row-major staging
  float* Wop   = alloc((size_t)DCAT * 64 * 4);     // W_out zero-padded to 64 cols
  float* WpkF  = alloc((size_t)NFEAT * DCAT * 4);  // fragment-packed B's
  float* WopF  = alloc((size_t)DCAT * 64 * 4);
  float* e1TF  = alloc((size_t)DCAT * NSTRUC * 4);
  float* e2TF  = alloc((size_t)NCLASS * NSTRUC * 4);
  float* ssrc  = alloc((size_t)NHEADS * NN * 4);
  float* sdst  = alloc((size_t)NHEADS * NN * 4);
  float* rows  = alloc((size_t)NHEADS * NN * 4);
  float* ho    = alloc((size_t)NN * NCLASS * 4);
  float* so_s  = alloc((size_t)NN * 4);
  float* so_d  = alloc((size_t)NN * 4);
  float* rowsO = alloc((size_t)NN * 4);
  float* aggO  = alloc((size_t)NN * NCLASS * 4);   // -> xo in place
  float* e1T   = alloc((size_t)DCAT * NSTRUC * 4);
  float* e2T   = alloc((size_t)NCLASS * NSTRUC * 4);

  const int BLK = 256;
  auto nblk = [](long long threads, int b) { return (unsigned)((threads + b - 1) / b); };

  // ---- zero accumulators + padded weight every call (graph replay safe) ----
  hipMemsetAsync(agg,   0, (size_t)NN * DCAT * 4,   stream);
  hipMemsetAsync(rows,  0, (size_t)NHEADS * NN * 4, stream);
  hipMemsetAsync(aggO,  0, (size_t)NN * NCLASS * 4, stream);
  hipMemsetAsync(rowsO, 0, (size_t)NN * 4,          stream);
  hipMemsetAsync(Wop,   0, (size_t)DCAT * 64 * 4,   stream);

  // ---- weight repack / pad / transposes, then fragment-pack each B ----
  repack_W<<<nblk((long long)NHEADS * NFEAT * NHID, BLK), BLK, 0, stream>>>(W_heads, Wpk);
  pad_Wout<<<nblk((long long)DCAT * NCLASS, BLK), BLK, 0, stream>>>(W_out, Wop);
  transpose_kn<<<nblk((long long)NSTRUC * DCAT, BLK), BLK, 0, stream>>>(enc1_w, e1T, NSTRUC, DCAT);
  transpose_kn<<<nblk((long long)NSTRUC * NCLASS, BLK), BLK, 0, stream>>>(enc2_w, e2T, NSTRUC, NCLASS);

  pack_frag<<<nblk((long long)NFEAT * DCAT, BLK), BLK, 0, stream>>>(Wpk, WpkF, NFEAT, DCAT);
  pack_frag<<<nblk((long long)DCAT * 64, BLK), BLK, 0, stream>>>(Wop, WopF, DCAT, 64);
  pack_frag<<<nblk((long long)DCAT * NSTRUC, BLK), BLK, 0, stream>>>(e1T, e1TF, DCAT, NSTRUC);
  pack_frag<<<nblk((long long)NCLASS * NSTRUC, BLK), BLK, 0, stream>>>(e2T, e2TF, NCLASS, NSTRUC);

  // ---- h_all = x @ W_all  [50000,256]@[256,512] (all heads fused, reads x once) ----
  {
    long long waves = (long long)(NN / 16) * (DCAT / 64);
    gemm_f32_wmma_n64<<<nblk(waves * 32, BLK), BLK, 0, stream>>>(
        x, WpkF, nullptr, h_all, NN, DCAT, DCAT, NFEAT, DCAT);
  }

  // ---- per-(node,head) scores ----
  head_scores<<<nblk((long long)NHEADS * NN * 32, BLK), BLK, 0, stream>>>(
      h_all, a_heads, ssrc, sdst);

  // ---- edge aggregation, head-by-head so the working set stays L2-resident ----
  for (int h = 0; h < NHEADS; ++h) {
    edge_head<<<nblk((long long)ECNT, BLK), BLK, 0, stream>>>(
        src, dst, ssrc + (size_t)h * NN, sdst + (size_t)h * NN,
        h_all, agg, rows + (size_t)h * NN, h);
  }

  // ---- hcat = elu(agg / rowsum)  (in place in agg) ----
  norm_heads<<<nblk((long long)NN * DCAT / 4, BLK), BLK, 0, stream>>>(agg, rows);

  // ---- output GAT layer: ho = hcat @ W_out  [50000,512]@[512,40] (B padded to 64) --
  {
    long long waves = (long long)(NN / 16) * 1;   // NnPad=64 -> one 64-col group
    gemm_f32_wmma_n64<<<nblk(waves * 32, BLK), BLK, 0, stream>>>(
        agg, WopF, nullptr, ho, NN, NCLASS, 64, DCAT, NCLASS);
  }
  out_scores<<<nblk((long long)NN * 32, BLK), BLK, 0, stream>>>(ho, a_out, so_s, so_d);
  edge_out<<<nblk((long long)ECNT, BLK), BLK, 0, stream>>>(
      src, dst, so_s, so_d, ho, aggO, rowsO);
  norm_out<<<nblk((long long)NN * NCLASS, BLK), BLK, 0, stream>>>(aggO, rowsO);  // aggO = xo

  // ---- z = xo @ enc2_w.T + b  [50000,40]@[40,128] ----
  {
    long long waves = (long long)(NN / 16) * (NSTRUC / 64);
    gemm_f32_wmma_n64<<<nblk(waves * 32, BLK), BLK, 0, stream>>>(
        aggO, e2TF, enc2_b, outz, NN, NSTRUC, NSTRUC, NCLASS, NSTRUC);
  }
  // ---- y = hcat @ enc1_w.T + b  [50000,512]@[512,128] ----
  {
    long long waves = (long long)(NN / 16) * (NSTRUC / 64);
    gemm_f32_wmma_n64<<<nblk(waves * 32, BLK), BLK, 0, stream>>>(
        agg, e1TF, enc1_b, outy, NN, NSTRUC, NSTRUC, DCAT, NSTRUC);
  }
  // ---- out0 = log_softmax(elu(xo)) ----
  logsoftmax40<<<nblk((long long)NN * 32, BLK), BLK, 0, stream>>>(aggO, out0);
}